// CombineInputsWithConstraints_6614249636256
// MI455X (gfx1250) — compile-verified
//
#include <hip/hip_runtime.h>
#include <cstdint>
#include <math.h>

#define KS     5
#define IH     720
#define IW     1280
#define NC     3
#define OH     716
#define OW     1276
#define NB     32

#define TH     16
#define TW     64
#define IN_H   (TH + 4)            // 20 input rows per tile
#define IN_WF  ((TW + 4) * NC)     // 204 floats per tile row (68 px * 3ch)
#define ROW_F  (IW * NC)           // 3840 floats per image row
#define TILE_F (IN_H * IN_WF)      // 4080
#define VS_F   (TH * IN_WF)        // 3264
#define HM_F   (TH * TW)           // 1024 homo flags
#define SMEM_F (TILE_F + 2 * VS_F + HM_F)  // 11632 floats = 46528 B
#define NA     ((size_t)OH * OW * NC)      // activations per batch (2,740,848; /4 exact)
#define NA4    (NA / 4)

typedef __attribute__((ext_vector_type(4))) float fvec4;   // native vector: OK for NT builtins

// ---------------- per-batch min/max reduction ----------------

__device__ __forceinline__ void atomicMinF(float* a, float v) {
  if (v >= 0.0f) atomicMin((int*)a, __float_as_int(v));
  else           atomicMax((unsigned int*)a, __float_as_uint(v));
}
__device__ __forceinline__ void atomicMaxF(float* a, float v) {
  if (v >= 0.0f) atomicMax((int*)a, __float_as_int(v));
  else           atomicMin((unsigned int*)a, __float_as_uint(v));
}

__global__ __launch_bounds__(64) void init_minmax_kernel(float* ws) {
  int t = threadIdx.x;
  if (t < NB) { ws[2 * t] = INFINITY; ws[2 * t + 1] = -INFINITY; }
}

__global__ __launch_bounds__(256) void reduce_minmax_kernel(const fvec4* __restrict__ act,
                                                            float* __restrict__ ws) {
  const int b = blockIdx.y;
  const fvec4* p = act + (size_t)b * NA4;
  float mn = INFINITY, mx = -INFINITY;
  for (size_t i = (size_t)blockIdx.x * 256 + threadIdx.x; i < NA4;
       i += (size_t)gridDim.x * 256) {
    fvec4 v = p[i];
    mn = fminf(mn, fminf(fminf(v[0], v[1]), fminf(v[2], v[3])));
    mx = fmaxf(mx, fmaxf(fmaxf(v[0], v[1]), fmaxf(v[2], v[3])));
  }
  __shared__ float smn[256];
  __shared__ float smx[256];
  smn[threadIdx.x] = mn;
  smx[threadIdx.x] = mx;
  __syncthreads();
  for (int s = 128; s > 0; s >>= 1) {
    if ((int)threadIdx.x < s) {
      smn[threadIdx.x] = fminf(smn[threadIdx.x], smn[threadIdx.x + s]);
      smx[threadIdx.x] = fmaxf(smx[threadIdx.x], smx[threadIdx.x + s]);
    }
    __syncthreads();
  }
  if (threadIdx.x == 0) {
    atomicMinF(&ws[2 * b],     smn[0]);
    atomicMaxF(&ws[2 * b + 1], smx[0]);
  }
}

// ---------------- fused stencil + blend kernel ----------------

__global__ __launch_bounds__(256) void fused_stencil_kernel(const float* __restrict__ in,
                                                            const float* __restrict__ act,
                                                            const float* __restrict__ mmws,
                                                            float* __restrict__ out) {
  extern __shared__ float smem[];
  float* tile  = smem;               // TILE_F: raw input tile (LDS byte offset 0)
  float* vs    = smem + TILE_F;      // VS_F: vertical 5-sums of x
  float* vq    = vs + VS_F;          // VS_F: vertical 5-sums of x^2
  float* hmask = vq + VS_F;          // HM_F: per-pixel homogeneity flag

  const int tid = threadIdx.x;
  const int b   = blockIdx.z;
  const int y0  = blockIdx.y * TH;
  const int x0  = blockIdx.x * TW;
  const int F0  = x0 * NC;                 // float-column of tile start in image row

  // ---- Stage A: stage the 2-D strided input tile into LDS ----
  // One Tensor-Data-Mover descriptor per block (20x204-float tile, row stride
  // 3840 floats). TDM OOB reads (past row end / past row 719) return zeros
  // that land only in halo slots consumed by skipped outputs.
#if __has_builtin(__builtin_amdgcn_tensor_load_to_lds)
  if (tid < 32) {                          // TDM is per-wave: issue from one wave only
    typedef unsigned int v4u __attribute__((ext_vector_type(4)));
    typedef int          v8i __attribute__((ext_vector_type(8)));
    typedef int          v4i __attribute__((ext_vector_type(4)));
    const uint64_t ga  = (uint64_t)(const void*)(in + ((size_t)(b * IH + y0) * ROW_F + F0));
    const unsigned td0 = (unsigned)(ROW_F - F0);   // tensor_dim0: floats left in row
    const unsigned td1 = (unsigned)(IH - y0);      // tensor_dim1: rows left in image
    // D# group 0: count=1 | lds_addr=0 | global_addr | type=2
    v4u g0 = { 1u,
               0u,
               (unsigned)(ga & 0xFFFFFFFFu),
               (unsigned)((ga >> 32) & 0x01FFFFFFu) | (2u << 30) };
    // D# group 1: data_size=4B; tensor_dim0/1; tile_dim0=204, tile_dim1=20,
    // tile_dim2=0; tensor_dim0_stride=3840; tensor_dim1_stride=0 (2-D)
    v8i g1 = { (int)(2u << 16),
               (int)((td0 & 0xFFFFu) << 16),
               (int)((td0 >> 16) | ((td1 & 0xFFFFu) << 16)),
               (int)((td1 >> 16) | ((unsigned)IN_WF << 16)),
               (int)(unsigned)IN_H,
               (int)(unsigned)ROW_F,
               0, 0 };
    v4i z4 = { 0, 0, 0, 0 };               // groups 2/3 unused (tile_dim2/3 = 0)
    v8i z8 = { 0, 0, 0, 0, 0, 0, 0, 0 };   // extra group (6-arg form): unused
    __builtin_amdgcn_tensor_load_to_lds(g0, g1, z4, z4, z8, 0);
  }
#if __has_builtin(__builtin_amdgcn_s_wait_tensorcnt)
  __builtin_amdgcn_s_wait_tensorcnt(0);
#else
  asm volatile("s_wait_tensorcnt 0x0" ::: "memory");
#endif
#else
  // Fallback: per-lane async b128 global->LDS copies.
  {
    const int fiMax = ROW_F - 4 - F0;      // clamp so b128 never crosses row end
    for (int i = tid; i < TILE_F / 4; i += 256) {
      const int r  = i / (IN_WF / 4);      // tile row (51 vec4 per row)
      const int v  = i - r * (IN_WF / 4);
      const int gy = (y0 + r < IH) ? (y0 + r) : (IH - 1);     // row clamp (edge tiles)
      int fi = v * 4;
      if (fi > fiMax) fi = fiMax;          // edge clamp: stay inside the buffer
      const uint64_t gaddr =
          (uint64_t)(const void*)(in + ((size_t)(b * IH + gy) * ROW_F + (F0 + fi)));
      const unsigned lds_off = (unsigned)((r * IN_WF + v * 4) * 4);
      asm volatile("global_load_async_to_lds_b128 %0, %1, off"
                   :
                   : "v"(lds_off), "v"(gaddr)
                   : "memory");
    }
    asm volatile("s_wait_asynccnt 0x0" ::: "memory");
  }
#endif
  __syncthreads();

  // ---- Stage B: vertical 5-row sums of x and x^2 (separable box filter) ----
  for (int i = tid; i < VS_F; i += 256) {
    const int r  = i / IN_WF;
    const int fc = i - r * IN_WF;
    const float* p = tile + r * IN_WF + fc;
    float s = 0.0f, q = 0.0f;
#pragma unroll
    for (int k = 0; k < KS; ++k) {
      const float x = p[k * IN_WF];
      s += x;
      q += x * x;
    }
    vs[i] = s;
    vq[i] = q;
  }
  __syncthreads();

  // ---- Stage C1: horizontal 5-sum -> mean/var/std -> per-pixel homo flag ----
  const float invN = 1.0f / 25.0f;
  for (int i = tid; i < TH * TW; i += 256) {
    const int r = i >> 6;
    const int x = i & 63;
    bool homo = true;
#pragma unroll
    for (int ch = 0; ch < NC; ++ch) {
      const int base = r * IN_WF + x * NC + ch;
      float s = 0.0f, q = 0.0f;
#pragma unroll
      for (int d = 0; d < KS; ++d) {
        s += vs[base + NC * d];
        q += vq[base + NC * d];
      }
      const float mean = s * invN;
      float var = q * invN - mean * mean;
      var = (var < 0.0f) ? 0.0f : var;
      const float sd = sqrtf(var);
      homo = homo && (sd >= 0.005f) && (sd <= 0.02f);
    }
    hmask[i] = homo ? 1.0f : 0.0f;
  }
  __syncthreads();

  // ---- Stage C2: float-granular, fully coalesced blend + store (vec4) ----
  const float mn   = mmws[2 * b];
  const float mx   = mmws[2 * b + 1];
  const float invR = 1.0f / (mx - mn);
  const int npx     = (OW - x0 < TW) ? (OW - x0) : TW;   // valid pixels this tile row
  const int fcLimit = npx * NC;                          // valid floats (192 or 180)

  for (int i = tid; i < TH * (TW * NC / 4); i += 256) {  // 16 * 48 vec4 slots
    const int r  = i / (TW * NC / 4);
    const int v  = i - r * (TW * NC / 4);
    const int fc = v * 4;
    const int oy = y0 + r;
    if (oy >= OH || fc >= fcLimit) continue;

    const size_t o = ((size_t)(b * OH + oy) * OW + x0) * NC + fc;
    const fvec4 a = *(const fvec4*)(act + o);

    fvec4 res;
#pragma unroll
    for (int e = 0; e < 4; ++e) {
      const int f  = fc + e;               // float index within the tile row
      const int x  = f / 3;                // pixel column (for homo lookup)
      const bool homo = hmask[r * TW + x] != 0.0f;
      const float cen = tile[(r + 2) * IN_WF + f + 6];   // (x+2)*3 + ch == f + 6
      res[e] = homo ? cen : (a[e] - mn) * invR;
    }
    __builtin_nontemporal_store(res, (fvec4*)(out + o));
  }
}

// ---------------- launch ----------------

extern "C" void kernel_launch(void* const* d_in, const int* in_sizes, int n_in,
                              void* d_out, int out_size, void* d_ws, size_t ws_size,
                              hipStream_t stream) {
  const float* cnn_inputs = (const float*)d_in[0];
  const float* constr     = (const float*)d_in[1];
  float* out = (float*)d_out;
  float* ws  = (float*)d_ws;   // 2 floats per batch: {min, max}

  init_minmax_kernel<<<1, 64, 0, stream>>>(ws);
  reduce_minmax_kernel<<<dim3(128, NB), 256, 0, stream>>>((const fvec4*)constr, ws);

  dim3 grid((OW + TW - 1) / TW, (OH + TH - 1) / TH, NB);  // 20 x 45 x 32
  fused_stencil_kernel<<<grid, 256, SMEM_F * sizeof(float), stream>>>(
      cnn_inputs, constr, ws, out);
}